// LocalEnergy_8761733284010
// MI455X (gfx1250) — compile-verified
//
#include <hip/hip_runtime.h>

typedef float v2f __attribute__((ext_vector_type(2)));
typedef float v8f __attribute__((ext_vector_type(8)));

#define N_MOL   4096
#define A_MAX   64
#define NF      512
#define NATOMS  (N_MOL * A_MAX)

// One block per molecule. 256 threads = 8 waves:
//   wave = fset*4 + tile ; fset in {0,1} selects features0/features1,
//   tile in {0..3} selects 16 atoms of the 64-atom molecule.
// Each wave runs 128 x V_WMMA_F32_16X16X4_F32 accumulations (two independent
// accumulator chains to let the matrix pipe overlap back-to-back WMMAs):
//   A (16x4 f32): lane L (L&15 = atom row M), half = L>>4 -> K pair {0,1} or {2,3}
//   B (4x16 f32): column N=0 = "energy" weight slice, N=1 = "propensity" slice,
//                 N>=2 = zeros (read from an LDS zero slot, stride 0).
// D (16x16 f32): col 0 = dot(F_row, we), col 1 = dot(F_row, wp).
__global__ __launch_bounds__(256) void local_energy_fused(
    const float* __restrict__ f0,  const float* __restrict__ f1,
    const float* __restrict__ we0, const float* __restrict__ we1,
    const float* __restrict__ be1, const float* __restrict__ wp0,
    const float* __restrict__ wp1,
    float* __restrict__ o_ce,   float* __restrict__ o_ae,
    float* __restrict__ o_ape,  float* __restrict__ o_prob,
    float* __restrict__ o_prop)
{
    __shared__ float sWE[2][NF];      // we0 / we1
    __shared__ float sWP[2][NF];      // wp0 / wp1
    __shared__ float sZero[4];        // feeds B columns N >= 2
    __shared__ float sPre[2][A_MAX];  // per-fset partial pre-energy
    __shared__ float sProp[2][A_MAX]; // per-fset partial propensity
    __shared__ float sRed[A_MAX];     // reduction scratch

    const int tid = threadIdx.x;
    const int mol = blockIdx.x;

    // Stage the four 512-float weight vectors into LDS (8 KB, read 128x each).
    for (int i = tid; i < NF; i += 256) {
        sWE[0][i] = we0[i];
        sWE[1][i] = we1[i];
        sWP[0][i] = wp0[i];
        sWP[1][i] = wp1[i];
    }
    if (tid < 4) sZero[tid] = 0.0f;
    __syncthreads();

    const int wave  = tid >> 5;
    const int lane  = tid & 31;
    const int laneN = lane & 15;   // A: row M ; B/D: column N
    const int half  = lane >> 4;   // 0 -> K{0,1}, M=v ; 1 -> K{2,3}, M=v+8
    const int koff  = half * 2;

    const int fset = wave >> 2;    // 0..1
    const int tile = wave & 3;     // 0..3 (16 atoms each)

    const float* F    = fset ? f1 : f0;
    const int    row  = mol * A_MAX + tile * 16 + laneN;
    const float* rowp = F + (size_t)row * NF + koff;   // 8B-aligned float2 stream

    const float* bptr  = (laneN == 0) ? &sWE[fset][koff]
                       : (laneN == 1) ? &sWP[fset][koff]
                       : &sZero[0];
    const int    bstep = (laneN < 2) ? 4 : 0;   // floats advanced per K-step

    v8f c0 = {};
    v8f c1 = {};
    #pragma unroll 4
    for (int j = 0; j < NF / 8; ++j) {
        // Two K-steps per trip on independent accumulators.
        v2f a0 = *(const v2f*)(rowp);
        v2f a1 = *(const v2f*)(rowp + 4);
        v2f b0, b1;
        b0.x = bptr[0];
        b0.y = bptr[1];
        b1.x = bptr[bstep];
        b1.y = bptr[bstep + 1];
        rowp += 8;
        bptr += 2 * bstep;
        // (neg_a, A, neg_b, B, c_mod, C, reuse_a, reuse_b)
        c0 = __builtin_amdgcn_wmma_f32_16x16x4_f32(false, a0, false, b0,
                                                   (short)0, c0, false, false);
        c1 = __builtin_amdgcn_wmma_f32_16x16x4_f32(false, a1, false, b1,
                                                   (short)0, c1, false, false);
    }
    const v8f c = c0 + c1;

    // D layout: lanes 0-15 hold M = v (VGPR v), lanes 16-31 hold M = v + 8.
    if (laneN < 2) {
        float* dst = (laneN == 0) ? &sPre[fset][tile * 16 + half * 8]
                                  : &sProp[fset][tile * 16 + half * 8];
        #pragma unroll
        for (int v = 0; v < 8; ++v) dst[v] = c[v];
    }
    __syncthreads();

    // ---- per-molecule softmax-weighted energy over the 64 packed atoms ----
    float pre = 0.0f, prop = 0.0f;
    if (tid < A_MAX) {
        pre  = sPre[0][tid] + sPre[1][tid] + be1[0];
        prop = sProp[0][tid] + sProp[1][tid];
        sRed[tid] = prop;
    }
    __syncthreads();
    if (tid < 32) sRed[tid] = fmaxf(sRed[tid], sRed[tid + 32]);
    __syncthreads();
    if (tid < 16) sRed[tid] = fmaxf(sRed[tid], sRed[tid + 16]);
    __syncthreads();
    if (tid <  8) sRed[tid] = fmaxf(sRed[tid], sRed[tid +  8]);
    __syncthreads();
    if (tid <  4) sRed[tid] = fmaxf(sRed[tid], sRed[tid +  4]);
    __syncthreads();
    if (tid <  2) sRed[tid] = fmaxf(sRed[tid], sRed[tid +  2]);
    __syncthreads();
    if (tid <  1) sRed[tid] = fmaxf(sRed[tid], sRed[tid +  1]);
    __syncthreads();
    const float m = sRed[0];
    __syncthreads();

    const float r = (tid < A_MAX) ? expf(prop - m) : 0.0f;
    if (tid < A_MAX) sRed[tid] = r;
    __syncthreads();
    if (tid < 32) sRed[tid] += sRed[tid + 32];
    __syncthreads();
    if (tid < 16) sRed[tid] += sRed[tid + 16];
    __syncthreads();
    if (tid <  8) sRed[tid] += sRed[tid +  8];
    __syncthreads();
    if (tid <  4) sRed[tid] += sRed[tid +  4];
    __syncthreads();
    if (tid <  2) sRed[tid] += sRed[tid +  2];
    __syncthreads();
    if (tid <  1) sRed[tid] += sRed[tid +  1];
    __syncthreads();
    const float z = sRed[0];
    __syncthreads();

    const float probv = r / z;
    const float ae    = probv * pre;
    if (tid < A_MAX) sRed[tid] = ae;
    __syncthreads();
    if (tid < 32) sRed[tid] += sRed[tid + 32];
    __syncthreads();
    if (tid < 16) sRed[tid] += sRed[tid + 16];
    __syncthreads();
    if (tid <  8) sRed[tid] += sRed[tid +  8];
    __syncthreads();
    if (tid <  4) sRed[tid] += sRed[tid +  4];
    __syncthreads();
    if (tid <  2) sRed[tid] += sRed[tid +  2];
    __syncthreads();
    if (tid <  1) sRed[tid] += sRed[tid +  1];
    __syncthreads();

    if (tid < A_MAX) {
        const int n = mol * A_MAX + tid;
        o_prop[n] = prop;
        o_ape[n]  = pre;
        o_prob[n] = probv;
        o_ae[n]   = ae;
    }
    if (tid == 0) o_ce[mol] = sRed[0];
}

extern "C" void kernel_launch(void* const* d_in, const int* in_sizes, int n_in,
                              void* d_out, int out_size, void* d_ws, size_t ws_size,
                              hipStream_t stream) {
    const float* f0  = (const float*)d_in[0];
    const float* f1  = (const float*)d_in[1];
    const float* we0 = (const float*)d_in[2];
    const float* we1 = (const float*)d_in[3];
    const float* be1 = (const float*)d_in[4];
    const float* wp0 = (const float*)d_in[5];
    const float* wp1 = (const float*)d_in[6];

    // d_out = (contributed_energy[4096], atom_energy[262144],
    //          atom_preenergy[262144], prob[262144], propensity[262144])
    float* out    = (float*)d_out;
    float* o_ce   = out;
    float* o_ae   = o_ce  + N_MOL;
    float* o_ape  = o_ae  + NATOMS;
    float* o_prob = o_ape + NATOMS;
    float* o_prop = o_prob + NATOMS;

    local_energy_fused<<<N_MOL, 256, 0, stream>>>(f0, f1, we0, we1, be1, wp0, wp1,
                                                  o_ce, o_ae, o_ape, o_prob, o_prop);
}